// ThrusterToHullConv_43293270344008
// MI455X (gfx1250) — compile-verified
//
#include <hip/hip_runtime.h>
#include <math.h>

typedef float v2f __attribute__((ext_vector_type(2)));
typedef float v4f __attribute__((ext_vector_type(4)));
typedef float v8f __attribute__((ext_vector_type(8)));

#define LN_EPS 1e-5f
#define T_DIM 32
#define E_DIM 16
#define H_DIM 32
#define HID   32
#define OUT   32
#define WAVES_PER_BLOCK 8

__global__ __launch_bounds__(256) void tth_conv_kernel(
    const float* __restrict__ x_src,
    const float* __restrict__ x_dst,
    const float* __restrict__ edge_attr,
    const long long* __restrict__ edge_index,
    const float* __restrict__ W1,
    const float* __restrict__ b1,
    const float* __restrict__ beta1,
    const float* __restrict__ ln_g,
    const float* __restrict__ ln_b,
    const float* __restrict__ W2,
    const float* __restrict__ b2,
    float* __restrict__ out,
    int E)
{
    // per-wave 16x32 staging tile (h between GEMMs, then output for coalesced store)
    __shared__ float lds_h[WAVES_PER_BLOCK][16 * 32];

    const int tid  = threadIdx.x;
    const int lane = tid & 31;
    const int wib  = tid >> 5;
    const int lrow = lane & 15;   // A-row / B,C-column within 16-wide tile
    const int half = lane >> 4;   // 0: K sub 0..1, 1: K sub 2..3
    const int q0   = 2 * half;

    float* hbuf = lds_h[wib];

    // ---- hoist W1 / W2 B-fragments into registers (f32 16x16x4 B layout) ----
    v2f w1f[20][2];
#pragma unroll
    for (int k = 0; k < 20; ++k) {
        const int q = 4 * k + q0;
#pragma unroll
        for (int t = 0; t < 2; ++t) {
            const int c = lrow + 16 * t;
            v2f b;
            b.x = W1[q * HID + c];
            b.y = W1[(q + 1) * HID + c];
            w1f[k][t] = b;
        }
    }
    v2f w2f[8][2];
#pragma unroll
    for (int k = 0; k < 8; ++k) {
        const int q = 4 * k + q0;
#pragma unroll
        for (int t = 0; t < 2; ++t) {
            const int c = lrow + 16 * t;
            v2f b;
            b.x = W2[q * OUT + c];
            b.y = W2[(q + 1) * OUT + c];
            w2f[k][t] = b;
        }
    }

    const float beta    = beta1[0];
    const float invBeta = 1.0f / beta;
    const float bias1_0 = b1[lrow], bias1_1 = b1[lrow + 16];
    const float g0  = ln_g[lrow],  g1  = ln_g[lrow + 16];
    const float be0 = ln_b[lrow],  be1 = ln_b[lrow + 16];
    const float bias2_0 = b2[lrow], bias2_1 = b2[lrow + 16];

    const int tilesTotal = (E + 15) >> 4;
    const int waveId = blockIdx.x * WAVES_PER_BLOCK + wib;
    const int nWaves = gridDim.x * WAVES_PER_BLOCK;

    for (int tile = waveId; tile < tilesTotal; tile += nWaves) {
        const int ebase = tile << 4;

        // prefetch next tile's edge_attr / edge_index streams
        const int nextTile = tile + nWaves;
        if (nextTile < tilesTotal) {
            const long long ne = (long long)(nextTile << 4) + lrow;
            __builtin_prefetch(edge_attr + ne * E_DIM, 0, 1);
            __builtin_prefetch(edge_index + ne, 0, 1);
        }

        // this lane supplies A-fragment data for edge row `lrow` of the tile
        int er = ebase + lrow;
        if (er >= E) er = E - 1;           // clamp: keep EXEC all-1s for WMMA
        const long long si = edge_index[er];
        const long long di = edge_index[(long long)E + er];
        const float* ps = x_src     + (long long)si * T_DIM;
        const float* pe = edge_attr + (long long)er * E_DIM;
        const float* pd = x_dst     + (long long)di * H_DIM;

        // ---- batch-load ALL 20 A-fragments first so loads pipeline (one wait,
        //      then 40 back-to-back WMMAs instead of load/wait/wmma serialization)
        v2f afrag[20];
#pragma unroll
        for (int k = 0; k < 8; ++k)        // x_src gather rows: L2-resident, cache normally
            afrag[k] = *(const v2f*)(ps + 4 * k + q0);
#pragma unroll
        for (int k = 8; k < 12; ++k)       // edge_attr: touched once -> non-temporal
            afrag[k] = __builtin_nontemporal_load((const v2f*)(pe + 4 * k - 32 + q0));
#pragma unroll
        for (int k = 12; k < 20; ++k)      // x_dst gather rows: L2-resident, cache normally
            afrag[k] = *(const v2f*)(pd + 4 * k - 48 + q0);

        // ---- GEMM1: [16 x 80] x [80 x 32], K in 20 chunks of 4 (f32 WMMA) ----
        v8f acc0 = {};
        v8f acc1 = {};
#pragma unroll
        for (int k = 0; k < 20; ++k) {
            acc0 = __builtin_amdgcn_wmma_f32_16x16x4_f32(false, afrag[k], false, w1f[k][0],
                                                         (short)0, acc0, false, false);
            acc1 = __builtin_amdgcn_wmma_f32_16x16x4_f32(false, afrag[k], false, w1f[k][1],
                                                         (short)0, acc1, false, false);
        }

        // ---- bias + AdaptiveSoftplus + LayerNorm (features live across 16 lanes) ----
        float h0[8], h1[8];
#pragma unroll
        for (int j = 0; j < 8; ++j) {
            float v0 = (acc0[j] + bias1_0) * beta;
            float v1 = (acc1[j] + bias1_1) * beta;
            // numerically stable softplus: max(x,0) + log1p(exp(-|x|))
            v0 = (fmaxf(v0, 0.0f) + log1pf(__expf(-fabsf(v0)))) * invBeta;
            v1 = (fmaxf(v1, 0.0f) + log1pf(__expf(-fabsf(v1)))) * invBeta;

            float s  = v0 + v1;
            float ss = v0 * v0 + v1 * v1;
#pragma unroll
            for (int off = 8; off >= 1; off >>= 1) {
                s  += __shfl_xor(s,  off, 32);
                ss += __shfl_xor(ss, off, 32);
            }
            const float mu  = s * (1.0f / 32.0f);
            const float var = ss * (1.0f / 32.0f) - mu * mu;
            const float rs  = rsqrtf(var + LN_EPS);
            h0[j] = (v0 - mu) * rs * g0 + be0;
            h1[j] = (v1 - mu) * rs * g1 + be1;
        }

        // ---- C-layout -> A-layout transpose via per-wave LDS tile ----
#pragma unroll
        for (int j = 0; j < 8; ++j) {
            const int r = j + 8 * half;
            hbuf[r * 32 + lrow]      = h0[j];
            hbuf[r * 32 + lrow + 16] = h1[j];
        }

        // ---- GEMM2: [16 x 32] x [32 x 32], K in 8 chunks of 4 ----
        v2f hfrag[8];
#pragma unroll
        for (int k = 0; k < 8; ++k)
            hfrag[k] = *(const v2f*)(hbuf + lrow * 32 + 4 * k + q0);

        v8f o0 = {};
        v8f o1 = {};
#pragma unroll
        for (int k = 0; k < 8; ++k) {
            o0 = __builtin_amdgcn_wmma_f32_16x16x4_f32(false, hfrag[k], false, w2f[k][0],
                                                       (short)0, o0, false, false);
            o1 = __builtin_amdgcn_wmma_f32_16x16x4_f32(false, hfrag[k], false, w2f[k][1],
                                                       (short)0, o1, false, false);
        }

        // ---- bias2, stage to LDS, coalesced non-temporal float4 global store ----
#pragma unroll
        for (int j = 0; j < 8; ++j) {
            const int r = j + 8 * half;
            hbuf[r * 32 + lrow]      = o0[j] + bias2_0;
            hbuf[r * 32 + lrow + 16] = o1[j] + bias2_1;
        }

        float* po = out + (long long)ebase * OUT;
        if (ebase + 16 <= E) {
#pragma unroll
            for (int i = 0; i < 4; ++i) {
                const int idx = lane * 4 + i * 128;
                __builtin_nontemporal_store(*(const v4f*)(hbuf + idx), (v4f*)(po + idx));
            }
        } else {
#pragma unroll
            for (int i = 0; i < 4; ++i) {
                const int idx = lane * 4 + i * 128;
                if (ebase + (idx >> 5) < E)
                    __builtin_nontemporal_store(*(const v4f*)(hbuf + idx), (v4f*)(po + idx));
            }
        }
    }
}

extern "C" void kernel_launch(void* const* d_in, const int* in_sizes, int n_in,
                              void* d_out, int out_size, void* d_ws, size_t ws_size,
                              hipStream_t stream) {
    const float*     x_src      = (const float*)d_in[0];
    const float*     x_dst      = (const float*)d_in[1];
    const float*     edge_attr  = (const float*)d_in[2];
    const long long* edge_index = (const long long*)d_in[3];
    const float*     W1         = (const float*)d_in[4];
    const float*     b1         = (const float*)d_in[5];
    const float*     beta1      = (const float*)d_in[6];
    const float*     ln_g       = (const float*)d_in[7];
    const float*     ln_b       = (const float*)d_in[8];
    const float*     W2         = (const float*)d_in[9];
    const float*     b2         = (const float*)d_in[10];
    float*           out        = (float*)d_out;

    const int E = in_sizes[2] / E_DIM;          // edge_attr is [E, 16]
    const int tiles = (E + 15) / 16;
    int grid = (tiles + WAVES_PER_BLOCK - 1) / WAVES_PER_BLOCK;
    if (grid > 2048) grid = 2048;               // persistent waves, grid-stride over tiles
    if (grid < 1) grid = 1;

    tth_conv_kernel<<<grid, 256, 0, stream>>>(x_src, x_dst, edge_attr, edge_index,
                                              W1, b1, beta1, ln_g, ln_b, W2, b2,
                                              out, E);
}